// GCN_41377714929856
// MI455X (gfx1250) — compile-verified
//
#include <hip/hip_runtime.h>

// GCN (3-layer GraphConv, norm='both') for MI455X / gfx1250.
// Dense GEMMs use V_WMMA_F32_16X16X4_F32 (wave32, one 16x16 tile per wave).
// Scatter phases are the real cost (~410MB gather + 410MB f32 atomics per
// layer, all L2-resident on the 192MB L2) and are laid out so 16 threads
// cooperatively stream one 256B edge row (float4 each), fully coalesced.

typedef __attribute__((ext_vector_type(2))) float v2f;
typedef __attribute__((ext_vector_type(8))) float v8f;

#define IN_F    128
#define HID_F   64
#define OUT_F   40
#define OUT_PAD 48   // pad N to 3 tiles of 16 for WMMA

// ---------------------------------------------------------------- utilities

__global__ void zero_f32(float* __restrict__ p, int n) {
  int t = blockIdx.x * blockDim.x + threadIdx.x;
  if (t < n) p[t] = 0.0f;
}

__global__ void count_degree(const int* __restrict__ idx, float* __restrict__ deg, int n) {
  int t = blockIdx.x * blockDim.x + threadIdx.x;
  if (t < n) atomicAdd(&deg[idx[t]], 1.0f);
}

__global__ void inv_sqrt_clamp(float* __restrict__ deg, int n) {
  int t = blockIdx.x * blockDim.x + threadIdx.x;
  if (t < n) deg[t] = rsqrtf(fmaxf(deg[t], 1.0f));
}

__global__ void pad_w3(const float* __restrict__ W3, float* __restrict__ W3p) {
  int t = blockIdx.x * blockDim.x + threadIdx.x;
  if (t >= HID_F * OUT_PAD) return;
  int r = t / OUT_PAD, c = t % OUT_PAD;
  W3p[t] = (c < OUT_F) ? W3[r * OUT_F + c] : 0.0f;
}

// ---------------------------------------------------------------- scatter

// 16 (WIDTH=64) or 12 (WIDTH=48) consecutive threads handle one edge;
// each loads one float4 of the source row and atomically adds it to dst row.
template <int WIDTH>
__global__ void scatter_add(const float* __restrict__ feat, float* __restrict__ agg,
                            const int* __restrict__ se, const int* __restrict__ de,
                            int nEdges) {
  constexpr unsigned CH = WIDTH / 4;
  unsigned t = blockIdx.x * blockDim.x + threadIdx.x;
  unsigned e = t / CH;
  unsigned ch = t - e * CH;
  if (e >= (unsigned)nEdges) return;
  int s = se[e], d = de[e];
  float4 v = ((const float4*)(feat + (size_t)s * WIDTH))[ch];
  float* dp = agg + (size_t)d * WIDTH + (size_t)ch * 4;
  atomicAdd(dp + 0, v.x);
  atomicAdd(dp + 1, v.y);
  atomicAdd(dp + 2, v.z);
  atomicAdd(dp + 3, v.w);
}

// ---------------------------------------------------------------- WMMA GEMMs
// A-matrix (16x4 f32) lane layout: lanes 0-15 rows M=0..15 with K={0,1},
// lanes 16-31 same rows with K={2,3}. C/D: VGPR j = row j (lanes 0-15) /
// row j+8 (lanes 16-31), col = lane&15.

// Layer 1: out = (X * norm_out[row]) @ W1   (K = 128, N = 64)
__global__ void gemm1_scaleA(const float* __restrict__ X, const float* __restrict__ W,
                             const float* __restrict__ norm_out, float* __restrict__ out,
                             int nNodes, int mTiles) {
  int wave = blockIdx.x * (blockDim.x >> 5) + (threadIdx.x >> 5);
  int lane = threadIdx.x & 31;
  int tm = wave >> 2;           // 4 N-tiles
  int tn = wave & 3;
  if (tm >= mTiles) return;     // wave-uniform: EXEC stays all-ones for WMMA
  int half = lane >> 4, l = lane & 15;
  int rowA = tm * 16 + l;
  int rA = rowA < nNodes ? rowA : nNodes - 1;
  float s = norm_out[rA];
  const float* xrow = X + (size_t)rA * IN_F;
  v8f c = {};
  for (int k0 = 0; k0 < IN_F; k0 += 4) {
    int ka = k0 + half * 2;
    float2 av = *(const float2*)(xrow + ka);
    v2f a; a.x = av.x * s; a.y = av.y * s;
    v2f b; b.x = W[ka * HID_F + tn * 16 + l];
           b.y = W[(ka + 1) * HID_F + tn * 16 + l];
    c = __builtin_amdgcn_wmma_f32_16x16x4_f32(false, a, false, b, (short)0, c, false, false);
  }
  int n = tn * 16 + l;
#pragma unroll
  for (int j = 0; j < 8; ++j) {
    int row = tm * 16 + half * 8 + j;
    if (row < nNodes) out[(size_t)row * HID_F + n] = c[j];
  }
}

// Layer 2: out = relu((A @ W2) * norm_in[row] + b2[n]) * norm_out[row]
// (norm_out pre-scale fused in for layer 3; K = 64, N = 64)
__global__ void gemm2_epi(const float* __restrict__ A, const float* __restrict__ W,
                          const float* __restrict__ bias, const float* __restrict__ norm_in,
                          const float* __restrict__ norm_out, float* __restrict__ out,
                          int nNodes, int mTiles) {
  int wave = blockIdx.x * (blockDim.x >> 5) + (threadIdx.x >> 5);
  int lane = threadIdx.x & 31;
  int tm = wave >> 2;
  int tn = wave & 3;
  if (tm >= mTiles) return;
  int half = lane >> 4, l = lane & 15;
  int rowA = tm * 16 + l;
  int rA = rowA < nNodes ? rowA : nNodes - 1;
  const float* arow = A + (size_t)rA * HID_F;
  v8f c = {};
  for (int k0 = 0; k0 < HID_F; k0 += 4) {
    int ka = k0 + half * 2;
    float2 av = *(const float2*)(arow + ka);
    v2f a; a.x = av.x; a.y = av.y;
    v2f b; b.x = W[ka * HID_F + tn * 16 + l];
           b.y = W[(ka + 1) * HID_F + tn * 16 + l];
    c = __builtin_amdgcn_wmma_f32_16x16x4_f32(false, a, false, b, (short)0, c, false, false);
  }
  int n = tn * 16 + l;
  float bv = bias[n];
#pragma unroll
  for (int j = 0; j < 8; ++j) {
    int row = tm * 16 + half * 8 + j;
    if (row < nNodes) {
      float v = c[j] * norm_in[row] + bv;
      v = fmaxf(v, 0.0f);
      out[(size_t)row * HID_F + n] = v * norm_out[row];
    }
  }
}

// Layer 3: out = A @ W3p   (A already norm_out-scaled; K = 64, N = 48 padded)
__global__ void gemm3_plain(const float* __restrict__ A, const float* __restrict__ W,
                            float* __restrict__ out, int nNodes, int mTiles) {
  int wave = blockIdx.x * (blockDim.x >> 5) + (threadIdx.x >> 5);
  int lane = threadIdx.x & 31;
  int tm = wave / 3;            // 3 N-tiles
  int tn = wave - tm * 3;
  if (tm >= mTiles) return;
  int half = lane >> 4, l = lane & 15;
  int rowA = tm * 16 + l;
  int rA = rowA < nNodes ? rowA : nNodes - 1;
  const float* arow = A + (size_t)rA * HID_F;
  v8f c = {};
  for (int k0 = 0; k0 < HID_F; k0 += 4) {
    int ka = k0 + half * 2;
    float2 av = *(const float2*)(arow + ka);
    v2f a; a.x = av.x; a.y = av.y;
    v2f b; b.x = W[ka * OUT_PAD + tn * 16 + l];
           b.y = W[(ka + 1) * OUT_PAD + tn * 16 + l];
    c = __builtin_amdgcn_wmma_f32_16x16x4_f32(false, a, false, b, (short)0, c, false, false);
  }
  int n = tn * 16 + l;
#pragma unroll
  for (int j = 0; j < 8; ++j) {
    int row = tm * 16 + half * 8 + j;
    if (row < nNodes) out[(size_t)row * OUT_PAD + n] = c[j];
  }
}

// ---------------------------------------------------------------- epilogues

// h1s = relu(agg * norm_in[i] + b1[f]) * norm_out[i]  (pre-scaled for layer 2)
__global__ void epilogue1(const float* __restrict__ agg, const float* __restrict__ norm_in,
                          const float* __restrict__ norm_out, const float* __restrict__ bias,
                          float* __restrict__ outp, int nNodes) {
  int t = blockIdx.x * blockDim.x + threadIdx.x;
  if (t >= nNodes * HID_F) return;
  int i = t >> 6;
  int f = t & (HID_F - 1);
  float v = agg[t] * norm_in[i] + bias[f];
  v = fmaxf(v, 0.0f);
  outp[t] = v * norm_out[i];
}

// out = agg48 * norm_in[i] + b3[f]   (final, no relu; drop pad columns)
__global__ void epilogue3(const float* __restrict__ agg48, const float* __restrict__ norm_in,
                          const float* __restrict__ bias, float* __restrict__ out, int nNodes) {
  int t = blockIdx.x * blockDim.x + threadIdx.x;
  if (t >= nNodes * OUT_F) return;
  int i = t / OUT_F;
  int f = t - i * OUT_F;
  out[t] = agg48[(size_t)i * OUT_PAD + f] * norm_in[i] + bias[f];
}

// ---------------------------------------------------------------- launch

static inline int cdiv_ll(long long a, long long b) { return (int)((a + b - 1) / b); }

extern "C" void kernel_launch(void* const* d_in, const int* in_sizes, int n_in,
                              void* d_out, int out_size, void* d_ws, size_t ws_size,
                              hipStream_t stream) {
  const float* X  = (const float*)d_in[0];
  const float* W1 = (const float*)d_in[1];
  const float* b1 = (const float*)d_in[2];
  const float* W2 = (const float*)d_in[3];
  const float* b2 = (const float*)d_in[4];
  const float* W3 = (const float*)d_in[5];
  const float* b3 = (const float*)d_in[6];
  const int* src  = (const int*)d_in[7];
  const int* dst  = (const int*)d_in[8];
  float* out = (float*)d_out;

  const int nNodes = in_sizes[0] / IN_F;   // 100000
  const int nEdges = in_sizes[7];          // 1600000
  const int mTiles = (nNodes + 15) / 16;

  // Workspace layout (~71 MB for 100k nodes)
  float* ws = (float*)d_ws;
  float* norm_out = ws;                         // nNodes   (deg_out -> rsqrt)
  float* norm_in  = ws + nNodes;                // nNodes   (deg_in  -> rsqrt)
  float* W3p      = ws + 2 * (size_t)nNodes;    // 64*48 padded W3
  size_t off = 2 * (size_t)nNodes + HID_F * OUT_PAD;
  off = (off + 15) & ~(size_t)15;
  float* bufA = (float*)d_ws + off;                 // nNodes*64
  float* bufB = bufA + (size_t)nNodes * HID_F;      // nNodes*64 (also agg48)
  float* bufC = bufB + (size_t)nNodes * HID_F;      // nNodes*48

  const dim3 blk(256);
  const int wavesH = mTiles * (HID_F / 16);    // 4 N-tiles
  const int wavesO = mTiles * (OUT_PAD / 16);  // 3 N-tiles

  // Degrees -> symmetric norms
  zero_f32<<<cdiv_ll(2LL * nNodes, 256), blk, 0, stream>>>(norm_out, 2 * nNodes);
  count_degree<<<cdiv_ll(nEdges, 256), blk, 0, stream>>>(src, norm_out, nEdges);
  count_degree<<<cdiv_ll(nEdges, 256), blk, 0, stream>>>(dst, norm_in, nEdges);
  inv_sqrt_clamp<<<cdiv_ll(2LL * nNodes, 256), blk, 0, stream>>>(norm_out, 2 * nNodes);
  pad_w3<<<cdiv_ll(HID_F * OUT_PAD, 256), blk, 0, stream>>>(W3, W3p);

  // ----- Layer 1: (X * norm_out) @ W1 -> scatter -> relu epi (pre-scale L2)
  gemm1_scaleA<<<cdiv_ll((long long)wavesH * 32, 256), blk, 0, stream>>>(
      X, W1, norm_out, bufA, nNodes, mTiles);
  zero_f32<<<cdiv_ll((long long)nNodes * HID_F, 256), blk, 0, stream>>>(bufB, nNodes * HID_F);
  scatter_add<HID_F><<<cdiv_ll((long long)nEdges * (HID_F / 4), 256), blk, 0, stream>>>(
      bufA, bufB, src, dst, nEdges);
  epilogue1<<<cdiv_ll((long long)nNodes * HID_F, 256), blk, 0, stream>>>(
      bufB, norm_in, norm_out, b1, bufA, nNodes);

  // ----- Layer 2: scatter -> (agg @ W2) with fused relu epi (pre-scale L3)
  zero_f32<<<cdiv_ll((long long)nNodes * HID_F, 256), blk, 0, stream>>>(bufB, nNodes * HID_F);
  scatter_add<HID_F><<<cdiv_ll((long long)nEdges * (HID_F / 4), 256), blk, 0, stream>>>(
      bufA, bufB, src, dst, nEdges);
  gemm2_epi<<<cdiv_ll((long long)wavesH * 32, 256), blk, 0, stream>>>(
      bufB, W2, b2, norm_in, norm_out, bufA, nNodes, mTiles);

  // ----- Layer 3: (h2 * norm_out) @ W3p -> scatter -> final epi
  gemm3_plain<<<cdiv_ll((long long)wavesO * 32, 256), blk, 0, stream>>>(
      bufA, W3p, bufC, nNodes, mTiles);
  zero_f32<<<cdiv_ll((long long)nNodes * OUT_PAD, 256), blk, 0, stream>>>(bufB, nNodes * OUT_PAD);
  scatter_add<OUT_PAD><<<cdiv_ll((long long)nEdges * (OUT_PAD / 4), 256), blk, 0, stream>>>(
      bufC, bufB, src, dst, nEdges);
  epilogue3<<<cdiv_ll((long long)nNodes * OUT_F, 256), blk, 0, stream>>>(
      bufB, norm_in, b3, out, nNodes);
}